// LSTM_6150393167882
// MI455X (gfx1250) — compile-verified
//
#include <hip/hip_runtime.h>
#include <math.h>

typedef __attribute__((ext_vector_type(2))) float v2f;
typedef __attribute__((ext_vector_type(8))) float v8f;
typedef __attribute__((ext_vector_type(4))) float f4v;
typedef int v4i __attribute__((vector_size(16)));   // matches builtin param type

#define BATCH   64
#define TSTEPS  512
#define HDIM    1024
#define G4H     4096   // 4*H
#define BT      (BATCH * TSTEPS)

// --------------------------------------------------------------------------
// CDNA5 async global->LDS path (ASYNCcnt-tracked, bypasses VGPRs).
// Builtin signature (from compiler diagnostic):
//   void __builtin_amdgcn_global_load_async_to_lds_b128(
//       v4i __attribute__((address_space(1)))* src,
//       v4i __attribute__((address_space(3)))* dst, imm int offset, imm int cpol)
// Guarded so the file also compiles on toolchains without the builtin.
// --------------------------------------------------------------------------
#if defined(__HIP_DEVICE_COMPILE__) && defined(__gfx1250__) && \
    __has_builtin(__builtin_amdgcn_global_load_async_to_lds_b128)
#define HAVE_ASYNC_LDS 1
typedef __attribute__((address_space(1))) v4i* as1_v4i;
typedef __attribute__((address_space(3))) v4i* as3_v4i;
#else
#define HAVE_ASYNC_LDS 0
#endif

__device__ __forceinline__ void cp_async16(const float* g, float* l) {
#if HAVE_ASYNC_LDS
    __builtin_amdgcn_global_load_async_to_lds_b128(
        (as1_v4i)(g), (as3_v4i)(l), /*imm offset=*/0, /*cpol=*/0);
#else
    *(f4v*)l = *(const f4v*)g;
#endif
}

__device__ __forceinline__ void wait_async_le8() {
#if HAVE_ASYNC_LDS
    asm volatile("s_wait_asynccnt 0x8" ::: "memory");   // stage i done, i+1 in flight
#endif
}
__device__ __forceinline__ void wait_async_0() {
#if HAVE_ASYNC_LDS
    asm volatile("s_wait_asynccnt 0x0" ::: "memory");
#endif
}

// ---------------------------------------------------------------------------
// Generic fp32 WMMA GEMM: C[M,N] = A[M,K] @ B[K,N]   (row-major, no bias)
// Requirements: M % 64 == 0, N % 64 == 0, K % 32 == 0.
// Block = 128 threads (4 waves, wave32). Tile 64x64, each wave owns 32x32
// (2x2 tiles of 16x16). K staged through double-buffered LDS in chunks of 32
// via GLOBAL_LOAD_ASYNC_TO_LDS_B128 (copy of stage i+1 overlaps WMMA of i).
// Uses V_WMMA_F32_16X16X4_F32 (full fp32 matrix path).
// ---------------------------------------------------------------------------
__global__ __launch_bounds__(128)
void lstm_gemm_f32_wmma(const float* __restrict__ A,
                        const float* __restrict__ B,
                        float* __restrict__ C,
                        int M, int N, int K) {
    // padded LDS tiles: A stride 36 (conflict-free 16-row column reads),
    // B stride 80 (half-wave K-rows hit disjoint bank ranges). Double-buffered.
    __shared__ float Ash[2][64 * 36];
    __shared__ float Bsh[2][32 * 80];

    const int tid   = threadIdx.x;
    const int lane  = tid & 31;
    const int wave  = tid >> 5;
    const int wm    = (wave >> 1) * 32;   // wave tile origin (m) inside 64x64
    const int wn    = (wave & 1) * 32;    // wave tile origin (n)
    const int lhalf = lane >> 4;          // 0: lanes 0-15, 1: lanes 16-31
    const int l16   = lane & 15;

    const int mTile = blockIdx.y * 64;
    const int nTile = blockIdx.x * 64;

    v8f acc[2][2];
    const v8f vzero = {0.f, 0.f, 0.f, 0.f, 0.f, 0.f, 0.f, 0.f};
    acc[0][0] = vzero; acc[0][1] = vzero; acc[1][0] = vzero; acc[1][1] = vzero;

    // issue one K-stage (32-deep panel) of async copies into LDS buffer `par`
    auto issue_stage = [&](int s, int par) {
        const int k0 = s << 5;
        // A tile: 64 rows x 32 k  (512 float4 -> 4 per thread)
#pragma unroll
        for (int j = 0; j < 4; ++j) {
            int q   = tid + j * 128;
            int row = q >> 3;
            int kq  = (q & 7) << 2;
            cp_async16(&A[(size_t)(mTile + row) * K + k0 + kq],
                       &Ash[par][row * 36 + kq]);
        }
        // B tile: 32 k-rows x 64 n (512 float4 -> 4 per thread)
#pragma unroll
        for (int j = 0; j < 4; ++j) {
            int q   = tid + j * 128;
            int row = q >> 4;
            int nq  = (q & 15) << 2;
            cp_async16(&B[(size_t)(k0 + row) * N + nTile + nq],
                       &Bsh[par][row * 80 + nq]);
        }
    };

    const int S = K >> 5;          // number of 32-deep K stages
    issue_stage(0, 0);

    for (int s = 0; s < S; ++s) {
        const int par = s & 1;
        if (s + 1 < S) {
            issue_stage(s + 1, par ^ 1);   // prefetch next stage (async, in flight)
            wait_async_le8();              // stage s landed (async completes in order)
        } else {
            wait_async_0();
        }
        __syncthreads();                   // publish stage s across waves

        const float* Ap = &Ash[par][0];
        const float* Bp = &Bsh[par][0];

        // ---- 8 WMMA k-steps of depth 4 over the staged 32-deep panel ----
#pragma unroll
        for (int ks = 0; ks < 8; ++ks) {
            // A 16x4 fp32 layout: lanes 0-15 hold {k,k+1}, lanes 16-31 {k+2,k+3}
            const int kk = ks * 4 + lhalf * 2;
            v2f a0 = *(const v2f*)&Ap[(wm + 0  + l16) * 36 + kk];
            v2f a1 = *(const v2f*)&Ap[(wm + 16 + l16) * 36 + kk];
            v2f b0, b1;
            b0.x = Bp[(kk    ) * 80 + wn + 0  + l16];
            b0.y = Bp[(kk + 1) * 80 + wn + 0  + l16];
            b1.x = Bp[(kk    ) * 80 + wn + 16 + l16];
            b1.y = Bp[(kk + 1) * 80 + wn + 16 + l16];

            acc[0][0] = __builtin_amdgcn_wmma_f32_16x16x4_f32(
                false, a0, false, b0, (short)0, acc[0][0], false, false);
            acc[0][1] = __builtin_amdgcn_wmma_f32_16x16x4_f32(
                false, a0, false, b1, (short)0, acc[0][1], false, false);
            acc[1][0] = __builtin_amdgcn_wmma_f32_16x16x4_f32(
                false, a1, false, b0, (short)0, acc[1][0], false, false);
            acc[1][1] = __builtin_amdgcn_wmma_f32_16x16x4_f32(
                false, a1, false, b1, (short)0, acc[1][1], false, false);
        }
        __syncthreads();                   // all waves done reading buf `par`
    }

    // ---- store: C/D layout -> vgpr r, lanes 0-15: M=r, lanes 16-31: M=r+8 ----
#pragma unroll
    for (int ms = 0; ms < 2; ++ms) {
#pragma unroll
        for (int ns = 0; ns < 2; ++ns) {
            const int mBase = mTile + wm + ms * 16 + lhalf * 8;
            const int nIdx  = nTile + wn + ns * 16 + l16;
#pragma unroll
            for (int r = 0; r < 8; ++r) {
                C[(size_t)(mBase + r) * N + nIdx] = acc[ms][ns][r];
            }
        }
    }
}

// ---------------------------------------------------------------------------
// Pointwise LSTM cell update for one timestep.
//   z = gates[b,t,:] + zbuf[b,:] + bias   (bias folded here, gate GEMMs have C=0)
//   c' = sig(f)*c + sig(i)*tanh(g);  h' = sig(o)*tanh(c')
// Writes h' into hstate + output sequence; at t==T-1 also writes final h/c.
// ---------------------------------------------------------------------------
__global__ __launch_bounds__(256)
void lstm_pointwise(const float* __restrict__ gates,  // [B,T,4H]
                    int t,
                    const float* __restrict__ zbuf,   // [B,4H]
                    const float* __restrict__ bias,   // [4H]
                    float* __restrict__ hstate,       // [B,H]
                    float* __restrict__ cstate,       // [B,H]
                    float* __restrict__ outseq,       // [B,T,H]
                    float* __restrict__ hT,           // [B,H] (finals)
                    float* __restrict__ cT,           // [B,H]
                    int writeFinals) {
    const int idx = blockIdx.x * blockDim.x + threadIdx.x;  // 0..B*H-1
    const int b   = idx >> 10;          // / HDIM
    const int hx  = idx & (HDIM - 1);

    const size_t grow = ((size_t)b * TSTEPS + t) * G4H;
    const size_t zrow = (size_t)b * G4H;

    const float zi = gates[grow + 0 * HDIM + hx] + zbuf[zrow + 0 * HDIM + hx] + bias[0 * HDIM + hx];
    const float zf = gates[grow + 1 * HDIM + hx] + zbuf[zrow + 1 * HDIM + hx] + bias[1 * HDIM + hx];
    const float zg = gates[grow + 2 * HDIM + hx] + zbuf[zrow + 2 * HDIM + hx] + bias[2 * HDIM + hx];
    const float zo = gates[grow + 3 * HDIM + hx] + zbuf[zrow + 3 * HDIM + hx] + bias[3 * HDIM + hx];

    const float ig = 1.0f / (1.0f + __expf(-zi));
    const float fg = 1.0f / (1.0f + __expf(-zf));
    const float gg = tanhf(zg);
    const float og = 1.0f / (1.0f + __expf(-zo));

    const float cn = fg * cstate[idx] + ig * gg;
    const float hn = og * tanhf(cn);

    cstate[idx] = cn;
    hstate[idx] = hn;
    outseq[((size_t)b * TSTEPS + t) * HDIM + hx] = hn;
    if (writeFinals) {
        hT[idx] = hn;
        cT[idx] = cn;
    }
}

// ---------------------------------------------------------------------------
// Host-side orchestration.
//  inputs: x[B,T,I], W[L,I,4H], U[L,H,4H], b[L,4H], h0[L,B,H], c0[L,B,H]
//  d_out : out[B,T,H] ++ hT[L,B,H] ++ cT[L,B,H]
// ---------------------------------------------------------------------------
extern "C" void kernel_launch(void* const* d_in, const int* in_sizes, int n_in,
                              void* d_out, int out_size, void* d_ws, size_t ws_size,
                              hipStream_t stream) {
    (void)in_sizes; (void)n_in; (void)out_size; (void)ws_size;

    const float* x  = (const float*)d_in[0];
    const float* W  = (const float*)d_in[1];
    const float* U  = (const float*)d_in[2];
    const float* bv = (const float*)d_in[3];
    const float* h0 = (const float*)d_in[4];
    const float* c0 = (const float*)d_in[5];

    float* out   = (float*)d_out;
    float* hTout = out + (size_t)BATCH * TSTEPS * HDIM;
    float* cTout = hTout + (size_t)2 * BATCH * HDIM;

    float* ws      = (float*)d_ws;
    float* gates   = ws;                                        // B*T*4H
    float* layerio = gates   + (size_t)BATCH * TSTEPS * G4H;    // B*T*H
    float* zbuf    = layerio + (size_t)BATCH * TSTEPS * HDIM;   // B*4H
    float* hstate  = zbuf    + (size_t)BATCH * G4H;             // B*H
    float* cstate  = hstate  + (size_t)BATCH * HDIM;            // B*H

    const dim3 gemmBlk(128);
    const dim3 bigGrid(G4H / 64, BT / 64);     // input-projection GEMM
    const dim3 stepGrid(G4H / 64, BATCH / 64); // recurrent GEMM (M=64)
    const int  pwBlocks = (BATCH * HDIM) / 256;

    for (int l = 0; l < 2; ++l) {
        const float* in = (l == 0) ? x : layerio;
        const float* Wl = W  + (size_t)l * HDIM * G4H;
        const float* Ul = U  + (size_t)l * HDIM * G4H;
        const float* bl = bv + (size_t)l * G4H;
        float* seqout   = (l == 1) ? out : layerio;

        // gates = in @ Wl     (bias added later in pointwise)
        lstm_gemm_f32_wmma<<<bigGrid, gemmBlk, 0, stream>>>(in, Wl, gates, BT, G4H, HDIM);

        (void)hipMemcpyAsync(hstate, h0 + (size_t)l * BATCH * HDIM,
                             (size_t)BATCH * HDIM * sizeof(float),
                             hipMemcpyDeviceToDevice, stream);
        (void)hipMemcpyAsync(cstate, c0 + (size_t)l * BATCH * HDIM,
                             (size_t)BATCH * HDIM * sizeof(float),
                             hipMemcpyDeviceToDevice, stream);

        for (int t = 0; t < TSTEPS; ++t) {
            // zbuf = hstate @ Ul
            lstm_gemm_f32_wmma<<<stepGrid, gemmBlk, 0, stream>>>(hstate, Ul, zbuf, BATCH, G4H, HDIM);
            const int fin = (t == TSTEPS - 1) ? 1 : 0;
            lstm_pointwise<<<pwBlocks, 256, 0, stream>>>(
                gates, t, zbuf, bl, hstate, cstate, seqout,
                hTout + (size_t)l * BATCH * HDIM,
                cTout + (size_t)l * BATCH * HDIM, fin);
        }
    }
}